// Normalizer_37323265803068
// MI455X (gfx1250) — compile-verified
//
#include <hip/hip_runtime.h>
#include <math.h>

typedef float v2f __attribute__((ext_vector_type(2)));
typedef float v4f __attribute__((ext_vector_type(4)));
typedef float v8f __attribute__((ext_vector_type(8)));
typedef int   v4i __attribute__((vector_size(4 * sizeof(int))));  // builtin's expected pointee

#if defined(__AMDGCN__) && __has_builtin(__builtin_amdgcn_global_load_async_to_lds_b128)
#define USE_ASYNC_LDS 1
#else
#define USE_ASYNC_LDS 0
#endif

#if defined(__AMDGCN__) && __has_builtin(__builtin_amdgcn_wmma_f32_16x16x4_f32)
#define USE_WMMA 1
#else
#define USE_WMMA 0
#endif

namespace {
constexpr int kN = 1024;       // row length (reference N)
constexpr int kRows = 16;      // rows per block (one 16-row slab)
constexpr int kPitch = kN + 4; // LDS row pitch in floats (pad: kills ds bank conflicts)
}

__global__ __launch_bounds__(256)
void rownorm_16x1024_kernel(const float* __restrict__ in0,
                            const float* __restrict__ in1,
                            float* __restrict__ out,
                            int slabs_per_input)
{
  __shared__ __attribute__((aligned(16))) float tile[kRows * kPitch];
  __shared__ float partials[kRows * 16];
  __shared__ float invbuf[kRows];

  const int t = (int)threadIdx.x;
  int sb = (int)blockIdx.x;

  const float* src;
  float* dst;
  if (sb < slabs_per_input) {
    src = in0;
    dst = out;
  } else {
    sb -= slabs_per_input;
    src = in1;
    dst = out + (size_t)slabs_per_input * kRows * kN;
  }
  const size_t base = (size_t)sb * (size_t)kRows * (size_t)kN;
  src += base;
  dst += base;

  // ---------- Phase 1: stage 16 rows (64 KB) into LDS, coalesced b128 ----------
#if USE_ASYNC_LDS
  #pragma unroll
  for (int i = 0; i < kRows; ++i) {
    // builtin signature (from compiler diagnostic): non-const AS(1) v4i* source
    __attribute__((address_space(1))) v4i* gp =
        (__attribute__((address_space(1))) v4i*)((void*)((const v4f*)(src + (size_t)i * kN) + t));
    __attribute__((address_space(3))) v4i* lp =
        (__attribute__((address_space(3))) v4i*)((void*)((v4f*)(tile + i * kPitch) + t));
    __builtin_amdgcn_global_load_async_to_lds_b128(gp, lp, 0, 0);
  }
  #if __has_builtin(__builtin_amdgcn_s_wait_asynccnt)
  __builtin_amdgcn_s_wait_asynccnt(0);
  #else
  asm volatile("s_wait_asynccnt 0" ::: "memory");
  #endif
#else
  #pragma unroll
  for (int i = 0; i < kRows; ++i) {
    v4f v = __builtin_nontemporal_load((const v4f*)(src + (size_t)i * kN) + t);
    *((v4f*)(tile + i * kPitch) + t) = v;
  }
#endif
  __syncthreads();

  // ---------- Phase 2: row sums -> inv (nan/inf -> 0), into invbuf[16] ----------
#if USE_WMMA
  // Wave 0 only: D = A(16x4) x ones(4x16) + C accumulates rowsums, replicated
  // across N.  A layout (ISA 7.12.2, 32-bit 16x4): lane m (and 16+m) holds row m;
  // VGPR0/1 = K {0,1} for lanes 0-15, K {2,3} for lanes 16-31.
  if (t < 32) {
    const int lane = t;
    const int m = lane & 15;
    const int koff = (lane >> 4) << 1;  // 0 or 2
    const float* rowp = tile + m * kPitch;
    v8f acc = {0.f, 0.f, 0.f, 0.f, 0.f, 0.f, 0.f, 0.f};
    const v2f bones = {1.0f, 1.0f};
    #pragma unroll 4
    for (int c = 0; c < kN; c += 4) {
      v2f a;
      a.x = rowp[c + koff];
      a.y = rowp[c + koff + 1];
      acc = __builtin_amdgcn_wmma_f32_16x16x4_f32(
          false, a, false, bones, (short)0, acc, false, false);
    }
    // D layout: VGPR r = row r (lanes 0-15) / row r+8 (lanes 16-31).
    if ((lane & 15) == 0) {  // lanes 0 and 16 extract 8 rows each
      union { v8f v; float f[8]; } u;
      u.v = acc;
      const int rbase = (lane >> 4) << 3;
      #pragma unroll
      for (int r = 0; r < 8; ++r) {
        float inv = 1.0f / u.f[r];
        invbuf[rbase + r] = __builtin_isfinite(inv) ? inv : 0.0f;
      }
    }
  }
  (void)partials;
#else
  {
    const int row = t >> 4, seg = t & 15;
    const float* p = tile + row * kPitch + seg * 64;
    float s = 0.0f;
    #pragma unroll 8
    for (int i = 0; i < 64; ++i) s += p[i];
    partials[t] = s;
  }
  __syncthreads();
  if (t < kRows) {
    float d = 0.0f;
    #pragma unroll
    for (int i = 0; i < 16; ++i) d += partials[t * 16 + i];
    float inv = 1.0f / d;
    invbuf[t] = __builtin_isfinite(inv) ? inv : 0.0f;
  }
#endif
  __syncthreads();

  // ---------- Phase 3: scale rows and stream out (non-temporal b128) ----------
  #pragma unroll
  for (int i = 0; i < kRows; ++i) {
    v4f v = *((const v4f*)(tile + i * kPitch) + t);
    v *= invbuf[i];
    __builtin_nontemporal_store(v, (v4f*)(dst + (size_t)i * kN) + t);
  }
}

extern "C" void kernel_launch(void* const* d_in, const int* in_sizes, int n_in,
                              void* d_out, int out_size, void* d_ws, size_t ws_size,
                              hipStream_t stream) {
  (void)n_in; (void)out_size; (void)d_ws; (void)ws_size;
  const float* a0 = (const float*)d_in[0];
  const float* a1 = (const float*)d_in[1];
  float* out = (float*)d_out;

  // in_sizes[0] = B*N*N with N = 1024 -> slabs of 16 rows per input
  const int slabs_per_input = in_sizes[0] / (kN * kRows);  // 16*1024*1024/(1024*16)=1024
  dim3 grid((unsigned)(2 * slabs_per_input));
  dim3 block(256);
  rownorm_16x1024_kernel<<<grid, block, 0, stream>>>(a0, a1, out, slabs_per_input);
}